// SPDNet_2_18373870092356
// MI455X (gfx1250) — compile-verified
//
#include <hip/hip_runtime.h>
#include <math.h>

// ---------------------------------------------------------------------------
// SPDNet forward for MI455X (gfx1250, wave32):
//   Kernel 1: one wave per 60x60 SPD matrix:
//     - LDS-stage X, compute Y = W X W^T (20x20)
//     - parallel-order Jacobi eigendecomposition (round-robin schedule,
//       10 disjoint rotations per round applied by 20 lanes in parallel)
//     - Z = U log(max(S,eps)) U^T, upper-tri vectorize with sqrt(2)
//   Kernel 2: head GEMM [8192,840]x[840,11]^T via V_WMMA_F32_16X16X4_F32,
//     with lw zero-padded to 16 rows in LDS so the inner loop carries no
//     EXEC-mask predication (WMMA requires EXEC all-1s anyway).
// ---------------------------------------------------------------------------

typedef __attribute__((ext_vector_type(2))) float v2f;
typedef __attribute__((ext_vector_type(8))) float v8f;

static constexpr int Bsz   = 8192;
static constexpr int CHn   = 4;
static constexpr int Cdim  = 60;
static constexpr int Odim  = 20;
static constexpr int Nmat  = Bsz * CHn;              // 32768
static constexpr int Fdim  = Odim * (Odim + 1) / 2;  // 210
static constexpr int FEAT  = CHn * Fdim;             // 840
static constexpr int NC    = 11;
static constexpr int NPAD  = 16;                     // N padded for WMMA
static constexpr int SWEEPS = 10;

#define EPS_CLAMP 1e-4f
#define SQRT2_F   1.4142135623730951f

__global__ __launch_bounds__(32)
void spd_branch_kernel(const float* __restrict__ x,
                       const float* __restrict__ W,
                       float* __restrict__ v)
{
    __shared__ float Xs[Cdim * Cdim];   // 3600
    __shared__ float Ws[Odim * Cdim];   // 1200
    __shared__ float Ts[Odim * Cdim];   // 1200
    __shared__ float Yr[Odim * Odim];   // 400
    __shared__ float As[Odim * Odim];   // 400
    __shared__ float Us[Odim * Odim];   // 400
    __shared__ float logsS[Odim];
    __shared__ float cR[10], sR[10];
    __shared__ int   pR[10], qR[10];

    const int lane = threadIdx.x;       // wave32: 0..31
    const int n    = blockIdx.x;        // matrix id in [0, 32768)
    const float* __restrict__ xm = x + (size_t)n * (Cdim * Cdim);

    for (int i = lane; i < Cdim * Cdim; i += 32) Xs[i] = xm[i];
    for (int i = lane; i < Odim * Cdim; i += 32) Ws[i] = W[i];
    __syncthreads();

    // T = W * X   (20 x 60)
    for (int e = lane; e < Odim * Cdim; e += 32) {
        const int o = e / Cdim, j = e % Cdim;
        float acc = 0.f;
        for (int i = 0; i < Cdim; ++i)
            acc = fmaf(Ws[o * Cdim + i], Xs[i * Cdim + j], acc);
        Ts[e] = acc;
    }
    __syncthreads();

    // Y = T * W^T  (20 x 20)
    for (int e = lane; e < Odim * Odim; e += 32) {
        const int o = e / Odim, p = e % Odim;
        float acc = 0.f;
        for (int j = 0; j < Cdim; ++j)
            acc = fmaf(Ts[o * Cdim + j], Ws[p * Cdim + j], acc);
        Yr[e] = acc;
    }
    __syncthreads();

    // symmetrize; U = I
    for (int e = lane; e < Odim * Odim; e += 32) {
        const int i = e / Odim, j = e % Odim;
        As[e] = 0.5f * (Yr[i * Odim + j] + Yr[j * Odim + i]);
        Us[e] = (i == j) ? 1.f : 0.f;
    }
    __syncthreads();

    // Parallel-order cyclic Jacobi: round-robin tournament over 20 indices:
    // 19 rounds x 10 disjoint (p,q) pairs, computed in closed form.
    for (int sweep = 0; sweep < SWEEPS; ++sweep) {
        for (int r = 0; r < 19; ++r) {
            if (lane < 10) {
                int p, q;
                if (lane == 0) { p = 19; q = r % 19; }
                else { p = (r + lane) % 19; q = (r - lane + 19) % 19; }
                if (p > q) { int tt = p; p = q; q = tt; }
                const float app = As[p * Odim + p];
                const float aqq = As[q * Odim + q];
                const float apq = As[p * Odim + q];
                float c = 1.f, s = 0.f;
                if (fabsf(apq) > 1e-12f) {
                    const float th = 0.5f * (aqq - app) / apq;
                    const float t  = copysignf(1.f, th) /
                                     (fabsf(th) + sqrtf(fmaf(th, th, 1.f)));
                    c = 1.f / sqrtf(fmaf(t, t, 1.f));
                    s = t * c;
                }
                pR[lane] = p; qR[lane] = q; cR[lane] = c; sR[lane] = s;
            }
            __syncthreads();
            // column updates of A (A <- A*J) and U (U <- U*J): lane k owns row k
            if (lane < Odim) {
                const int k = lane;
                #pragma unroll
                for (int t = 0; t < 10; ++t) {
                    const int p = pR[t], q = qR[t];
                    const float c = cR[t], s = sR[t];
                    const float akp = As[k * Odim + p], akq = As[k * Odim + q];
                    As[k * Odim + p] = fmaf(c, akp, -s * akq);
                    As[k * Odim + q] = fmaf(s, akp,  c * akq);
                    const float ukp = Us[k * Odim + p], ukq = Us[k * Odim + q];
                    Us[k * Odim + p] = fmaf(c, ukp, -s * ukq);
                    Us[k * Odim + q] = fmaf(s, ukp,  c * ukq);
                }
            }
            __syncthreads();
            // row updates of A (A <- J^T*A): lane k owns column k
            if (lane < Odim) {
                const int k = lane;
                #pragma unroll
                for (int t = 0; t < 10; ++t) {
                    const int p = pR[t], q = qR[t];
                    const float c = cR[t], s = sR[t];
                    const float apk = As[p * Odim + k], aqk = As[q * Odim + k];
                    As[p * Odim + k] = fmaf(c, apk, -s * aqk);
                    As[q * Odim + k] = fmaf(s, apk,  c * aqk);
                }
            }
            __syncthreads();
        }
    }

    if (lane < Odim)
        logsS[lane] = logf(fmaxf(As[lane * Odim + lane], EPS_CLAMP));
    __syncthreads();

    // tangent vectorization -> v[b][ch*210 + idx]
    const int b = n >> 2, ch = n & 3;
    float* __restrict__ vout = v + (size_t)b * FEAT + ch * Fdim;
    for (int idx = lane; idx < Fdim; idx += 32) {
        int i = 0, rem = idx;                 // row-major upper-tri index map
        while (rem >= Odim - i) { rem -= (Odim - i); ++i; }
        const int j = i + rem;
        float acc = 0.f;
        #pragma unroll
        for (int k = 0; k < Odim; ++k)
            acc = fmaf(Us[i * Odim + k] * logsS[k], Us[j * Odim + k], acc);
        vout[idx] = (i == j) ? acc : acc * SQRT2_F;
    }
}

// ---------------------------------------------------------------------------
// Head GEMM: out[8192,11] = v[8192,840] @ lw^T + lb, via f32 WMMA 16x16x4.
// One wave per 16-row tile; N padded 11 -> 16 with zero rows in LDS so the
// inner loop has NO predication (no exec save/restore around ds_load).
// A-fragment layout (ISA 32-bit A 16x4): lanes 0-15 K={k0,k0+1},
// lanes 16-31 K={k0+2,k0+3}, M = lane&15.
// B-fragment (4x16, rows striped across VGPR+lane-half), C/D in 8 VGPRs.
// ---------------------------------------------------------------------------
__global__ __launch_bounds__(32)
void head_gemm_wmma(const float* __restrict__ v,
                    const float* __restrict__ lw,
                    const float* __restrict__ lb,
                    float* __restrict__ out)
{
    __shared__ float lwS[NPAD * FEAT];  // 16*840 floats, rows 11..15 = 0
    __shared__ float lbS[NPAD];

    const int lane = threadIdx.x;
    for (int i = lane; i < NPAD * FEAT; i += 32)
        lwS[i] = (i < NC * FEAT) ? lw[i] : 0.f;
    if (lane < NPAD) lbS[lane] = (lane < NC) ? lb[lane] : 0.f;
    __syncthreads();

    const int m0   = blockIdx.x * 16;
    const int mrow = m0 + (lane & 15);
    const int half = lane >> 4;        // 0 or 1
    const int ncol = lane & 15;

    v8f acc = {};
    const float* __restrict__ vrow = v + (size_t)mrow * FEAT + 2 * half;
    const float* __restrict__ brow = lwS + ncol * FEAT + 2 * half;

    for (int k0 = 0; k0 < FEAT; k0 += 4) {
        v2f a, bf;
        a.x  = vrow[k0];
        a.y  = vrow[k0 + 1];
        bf.x = brow[k0];
        bf.y = brow[k0 + 1];
        // D = A(16x4) * B(4x16) + C ; all f32
        acc = __builtin_amdgcn_wmma_f32_16x16x4_f32(
            /*neg_a=*/false, a, /*neg_b=*/false, bf,
            /*c_mod=*/(short)0, acc, /*reuse_a=*/false, /*reuse_b=*/false);
    }

    if (ncol < NC) {
        #pragma unroll
        for (int vg = 0; vg < 8; ++vg) {
            const int row = m0 + vg + 8 * half;   // C/D layout: M = vg + 8*half
            out[row * NC + ncol] = acc[vg] + lbS[ncol];
        }
    }
}

extern "C" void kernel_launch(void* const* d_in, const int* in_sizes, int n_in,
                              void* d_out, int out_size, void* d_ws, size_t ws_size,
                              hipStream_t stream) {
    (void)in_sizes; (void)n_in; (void)out_size; (void)ws_size;
    const float* x  = (const float*)d_in[0];   // [8192,4,60,60]
    const float* W  = (const float*)d_in[1];   // [20,60]
    const float* lw = (const float*)d_in[2];   // [11,840]
    const float* lb = (const float*)d_in[3];   // [11]
    float* out = (float*)d_out;                // [8192,11]
    float* v   = (float*)d_ws;                 // needs 32768*210*4 = 27.5 MB

    spd_branch_kernel<<<Nmat, 32, 0, stream>>>(x, W, v);
    head_gemm_wmma<<<Bsz / 16, 32, 0, stream>>>(v, lw, lb, out);
}